// EuclideanVisitEncoder_62852551410334
// MI455X (gfx1250) — compile-verified
//
#include <hip/hip_runtime.h>
#include <stdint.h>

// ---------------------------------------------------------------------------
// Embedding-bag mean pool (pad_idx = 0)
//   code_ids  : [B, 64]  int32
//   emb       : [100000, 16] f32   (6.4 MB -> L2 resident on MI455X)
//   out       : [B, 16]  f32
//
// Memory-bound gather problem (819 MB of L2 gathers, ~70 MB HBM). WMMA is
// inapplicable (no shared matrix operand); the CDNA5-relevant path is the
// async global->LDS copy (ASYNCcnt) used to stage the index tile.
// ---------------------------------------------------------------------------

#ifndef USE_ASYNC_LDS
#define USE_ASYNC_LDS 1
#endif

#define THREADS        256
#define ROWS_PER_BLOCK 64     // 4 lanes per row
#define SEQ_L          64
#define DIM            16
#define LDS_STRIDE     65     // pad 64->65 words: quads in a wave hit distinct banks

__global__ __launch_bounds__(THREADS)
void pooled_emb_mean_kernel(const int* __restrict__ code_ids,
                            const float* __restrict__ emb,
                            float* __restrict__ out,
                            int Brows)
{
    __shared__ int s_ids[ROWS_PER_BLOCK * LDS_STRIDE];   // 16.6 KB of 320 KB

    const int tid  = threadIdx.x;
    const int r    = tid >> 2;       // local row 0..63 (staging row == gather row)
    const int sub  = tid & 3;        // quad lane: which float4 of the 16-float dim
    const int row0 = blockIdx.x * ROWS_PER_BLOCK;

    // ---------------- stage this block's 64x64 indices into LDS ----------------
    if (row0 + ROWS_PER_BLOCK <= Brows) {
#if USE_ASYNC_LDS
        // Each lane issues 4 async 16B copies: global (coalesced) -> padded LDS.
        #pragma unroll
        for (int k = sub; k < 16; k += 4) {
            const uint64_t gaddr =
                (uint64_t)(uintptr_t)(code_ids + (size_t)(row0 + r) * SEQ_L + k * 4);
            // low 32 bits of a generic shared pointer == LDS byte offset (ISA 10.2)
            const uint32_t loff =
                (uint32_t)(uintptr_t)&s_ids[r * LDS_STRIDE + k * 4];
            asm volatile("global_load_async_to_lds_b128 %0, %1, off"
                         :: "v"(loff), "v"(gaddr)
                         : "memory");
        }
        asm volatile("s_wait_asynccnt 0" ::: "memory");
#else
        #pragma unroll
        for (int k = sub; k < 16; k += 4) {
            const int4 v = *(const int4*)(code_ids + (size_t)(row0 + r) * SEQ_L + k * 4);
            int* dst = &s_ids[r * LDS_STRIDE + k * 4];
            dst[0] = v.x; dst[1] = v.y; dst[2] = v.z; dst[3] = v.w;
        }
#endif
    } else {
        // tail tile (not hit for B=200000, kept for generality): guarded scalar fill
        for (int k = sub; k < 16; k += 4) {
            #pragma unroll
            for (int j = 0; j < 4; ++j) {
                const int rr = row0 + r;
                int id = 0;
                if (rr < Brows) id = code_ids[(size_t)rr * SEQ_L + k * 4 + j];
                s_ids[r * LDS_STRIDE + k * 4 + j] = id;
            }
        }
    }
    __syncthreads();

    const int row = row0 + r;
    if (row >= Brows) return;

    // ---------------- gather + masked accumulate ----------------
    const int* ids = &s_ids[r * LDS_STRIDE];   // same addr within quad -> LDS broadcast
    float4 acc = make_float4(0.f, 0.f, 0.f, 0.f);
    int count = 0;

    #pragma unroll 8
    for (int t = 0; t < SEQ_L; ++t) {
        const int id = ids[t];
        // unconditional b128 gather (keeps loads pipelined, no EXEC divergence);
        // pad rows are masked arithmetically.
        const float4 e = *(const float4*)(emb + (size_t)id * DIM + sub * 4);
        const bool nz = (id != 0);
        const float m = nz ? 1.0f : 0.0f;
        acc.x += m * e.x;
        acc.y += m * e.y;
        acc.z += m * e.z;
        acc.w += m * e.w;
        count += nz ? 1 : 0;
    }

    const float denom = (count > 0) ? (float)count : 1.0f;
    const float inv   = 1.0f / denom;

    float4 res;
    res.x = acc.x * inv;
    res.y = acc.y * inv;
    res.z = acc.z * inv;
    res.w = acc.w * inv;

    *(float4*)(out + (size_t)row * DIM + sub * 4) = res;   // quad stores 64B contiguous
}

// ---------------------------------------------------------------------------
extern "C" void kernel_launch(void* const* d_in, const int* in_sizes, int n_in,
                              void* d_out, int out_size, void* d_ws, size_t ws_size,
                              hipStream_t stream)
{
    (void)n_in; (void)d_ws; (void)ws_size; (void)out_size;

    const int*   code_ids = (const int*)  d_in[0];   // [B, 64] int32
    const float* emb      = (const float*)d_in[1];   // [100000, 16] f32
    float*       out      = (float*)      d_out;     // [B, 16] f32

    const int Brows = in_sizes[0] / SEQ_L;           // 200000
    const int grid  = (Brows + ROWS_PER_BLOCK - 1) / ROWS_PER_BLOCK;

    pooled_emb_mean_kernel<<<grid, THREADS, 0, stream>>>(code_ids, emb, out, Brows);
}